// TensorDecompositionNetwork_65876208386287
// MI455X (gfx1250) — compile-verified
//
#include <hip/hip_runtime.h>
#include <hip/hip_bf16.h>

#define NN 10000
#define EE 100000
#define LL 9
#define HH 8
#define WN 576            // R*C
#define NL 90000          // N*L
#define WSTR 580          // padded 576 (bank-conflict-free: 580%64==4)
#define HSTR 68           // padded 64

typedef __attribute__((ext_vector_type(2))) float v2f;
typedef __attribute__((ext_vector_type(8))) float v8f;

__device__ __forceinline__ v8f wmma4(v2f a, v2f b, v8f c) {
  // V_WMMA_F32_16X16X4_F32 : D = A(16x4) * B(4x16) + C(16x16)
  return __builtin_amdgcn_wmma_f32_16x16x4_f32(false, a, false, b, (short)0, c, false, false);
}

// ---------------------------------------------------------------------------
// Kernel 1: src_feat = X @ W_src (+b_src on l==0 rows), dst_feat = X @ W_dst
// One 16-row tile per block; 8 waves cover 8 col-tiles (4 src + 4 dst).
// ---------------------------------------------------------------------------
__global__ void __launch_bounds__(256) k_node_lin(
    const float* __restrict__ X, const float* __restrict__ Wsrc,
    const float* __restrict__ bsrc, const float* __restrict__ Wdst,
    float* __restrict__ srcF, float* __restrict__ dstF)
{
  __shared__ float At[16 * HSTR];
  const int t = threadIdx.x;
  const int wave = t >> 5, lane = t & 31;
  const int m0 = blockIdx.x * 16;
  {
    float4 v = ((const float4*)(X + (size_t)m0 * 64))[t];  // 256 float4 = 16x64
    int row = (t * 4) >> 6, col = (t * 4) & 63;
    float* p = &At[row * HSTR + col];
    p[0] = v.x; p[1] = v.y; p[2] = v.z; p[3] = v.w;
  }
  __syncthreads();
  const float* W = (wave < 4) ? Wsrc : Wdst;
  const int n0 = (wave & 3) * 16;
  const int mrow = lane & 15, hi = lane >> 4, koff = hi * 2;
  v8f acc = {};
  for (int k0 = 0; k0 < 64; k0 += 4) {
    v2f a, b;
    a.x = At[mrow * HSTR + k0 + koff];
    a.y = At[mrow * HSTR + k0 + koff + 1];
    b.x = W[(size_t)(k0 + koff) * 64 + n0 + mrow];
    b.y = W[(size_t)(k0 + koff + 1) * 64 + n0 + mrow];
    acc = wmma4(a, b, acc);
  }
  float* out = (wave < 4) ? srcF : dstF;
  const bool addb = (wave < 4);
#pragma unroll
  for (int g = 0; g < 8; ++g) {
    int row = m0 + g + hi * 8;
    int col = n0 + mrow;
    float v = acc[g];
    if (addb && (row % LL) == 0) v += bsrc[col];
    out[(size_t)row * 64 + col] = v;
  }
}

// ---------------------------------------------------------------------------
// Shared per-edge-group stage: radial MLP -> w, msg gather, CP-TP1 (in place).
// One wave handles 16 edges. All big buffers in (dynamic) LDS.
// cc layout: [0]=U1 [81]=V1 [162]=Y1 [243]=U2 [324]=V2 [405]=Y2
// ---------------------------------------------------------------------------
__device__ __forceinline__ void edge_common(
    int eg0, int lane,
    const float* __restrict__ ea, const float* __restrict__ es,
    const float* __restrict__ rw1, const float* __restrict__ rb1,
    const float* __restrict__ lnw, const float* __restrict__ lnb,
    const float* __restrict__ rw2, const float* __restrict__ roff,
    const float* __restrict__ U1, const float* __restrict__ V1,
    const float* __restrict__ Y1, const float* __restrict__ U2,
    const float* __restrict__ V2, const float* __restrict__ Y2,
    const float* __restrict__ srcF, const float* __restrict__ dstF,
    const int* __restrict__ esrc, const int* __restrict__ edst,
    float* hb, float* wb, float* mb, float* shb, float* ys1b, float* cc)
{
  const int mrow = lane & 15, hi = lane >> 4, koff = hi * 2;

  if (eg0 + 16 < EE) {  // prefetch next tile's scalars (global_prefetch)
    __builtin_prefetch(es + (size_t)(eg0 + 16) * 64, 0, 1);
    __builtin_prefetch(ea + (size_t)(eg0 + 16) * LL, 0, 1);
  }
  for (int i = lane; i < 81; i += 32) {
    cc[i] = U1[i]; cc[81 + i] = V1[i]; cc[162 + i] = Y1[i];
    cc[243 + i] = U2[i]; cc[324 + i] = V2[i]; cc[405 + i] = Y2[i];
  }
  for (int i = lane; i < 256; i += 32) {          // edge_scalars 16x64 -> hb
    float4 v = ((const float4*)(es + (size_t)eg0 * 64))[i];
    int row = (i * 4) >> 6, col = (i * 4) & 63;
    float* p = &hb[row * HSTR + col];
    p[0] = v.x; p[1] = v.y; p[2] = v.z; p[3] = v.w;
  }
  for (int i = lane; i < 16 * LL; i += 32)        // edge_attr -> shb
    shb[(i / LL) * 12 + (i % LL)] = ea[(size_t)(eg0 + i / LL) * LL + (i % LL)];
  for (int e = 0; e < 16; ++e) {                  // msg gather (L2-resident)
    int se = esrc[eg0 + e], de = edst[eg0 + e];
    const float4* sp = (const float4*)(srcF + (size_t)se * WN);
    const float4* dp = (const float4*)(dstF + (size_t)de * WN);
    for (int i = lane; i < 144; i += 32) {
      float4 a4 = sp[i], b4 = dp[i];
      float* o = &mb[e * WSTR + i * 4];
      o[0] = a4.x + b4.x; o[1] = a4.y + b4.y;
      o[2] = a4.z + b4.z; o[3] = a4.w + b4.w;
    }
  }
  __syncthreads();

  // h = es @ rad_w1  (16x64, 4 WMMA col-tiles x 16 k-steps)
  v8f hacc[4] = {};
  for (int k0 = 0; k0 < 64; k0 += 4) {
    v2f a;
    a.x = hb[mrow * HSTR + k0 + koff];
    a.y = hb[mrow * HSTR + k0 + koff + 1];
#pragma unroll
    for (int nt = 0; nt < 4; ++nt) {
      v2f b;
      b.x = rw1[(size_t)(k0 + koff) * 64 + nt * 16 + mrow];
      b.y = rw1[(size_t)(k0 + koff + 1) * 64 + nt * 16 + mrow];
      hacc[nt] = wmma4(a, b, hacc[nt]);
    }
  }
  __syncthreads();
#pragma unroll
  for (int nt = 0; nt < 4; ++nt)
#pragma unroll
    for (int g = 0; g < 8; ++g)
      hb[(g + hi * 8) * HSTR + nt * 16 + mrow] = hacc[nt][g] + rb1[nt * 16 + mrow];
  __syncthreads();

  // LayerNorm + affine + SiLU (lane pair per row, shfl-combine halves)
  {
    int row = mrow, cb = hi * 32;
    float s = 0.f, s2 = 0.f;
    for (int c = 0; c < 32; ++c) {
      float v = hb[row * HSTR + cb + c];
      s += v; s2 += v * v;
    }
    s += __shfl_xor(s, 16);
    s2 += __shfl_xor(s2, 16);
    float mean = s * (1.f / 64.f);
    float var = s2 * (1.f / 64.f) - mean * mean;
    float rstd = rsqrtf(var + 1e-5f);
    for (int c = 0; c < 32; ++c) {
      int col = cb + c;
      float v = (hb[row * HSTR + col] - mean) * rstd;
      v = v * lnw[col] + lnb[col];
      hb[row * HSTR + col] = v / (1.f + expf(-v));     // silu
    }
  }
  __syncthreads();

  // w = h @ rad_w2 + offset  (16x576, 36 WMMA tiles)
  for (int nt = 0; nt < 36; ++nt) {
    v8f acc = {};
    for (int k0 = 0; k0 < 64; k0 += 4) {
      v2f a, b;
      a.x = hb[mrow * HSTR + k0 + koff];
      a.y = hb[mrow * HSTR + k0 + koff + 1];
      b.x = rw2[(size_t)(k0 + koff) * WN + nt * 16 + mrow];
      b.y = rw2[(size_t)(k0 + koff + 1) * WN + nt * 16 + mrow];
      acc = wmma4(a, b, acc);
    }
#pragma unroll
    for (int g = 0; g < 8; ++g)
      wb[(g + hi * 8) * WSTR + nt * 16 + mrow] = acc[g] + roff[nt * 16 + mrow];
  }
  // ys1[e][r] = sum_l sh[e][l] * Y1[l][r]
  for (int i = lane; i < 144; i += 32) {
    int e = i / 9, r = i % 9;
    float s = 0.f;
#pragma unroll
    for (int l = 0; l < 9; ++l) s += shb[e * 12 + l] * cc[162 + l * 9 + r];
    ys1b[e * 12 + r] = s;
  }
  __syncthreads();

  // CP-TP1 in place on mb (per-(e,c) 9x9 contractions)
  {
    int e = mrow, cb = hi * 32;
    for (int c = 0; c < 32; ++c) {
      int col = cb + c;
      float m[9], xv[9];
#pragma unroll
      for (int l = 0; l < 9; ++l) m[l] = mb[e * WSTR + l * 64 + col];
#pragma unroll
      for (int r = 0; r < 9; ++r) {
        float s = 0.f;
#pragma unroll
        for (int l = 0; l < 9; ++l) s += m[l] * cc[81 + l * 9 + r];
        xv[r] = s * ys1b[e * 12 + r] * wb[e * WSTR + r * 64 + col];
      }
#pragma unroll
      for (int l = 0; l < 9; ++l) {
        float s = 0.f;
#pragma unroll
        for (int r = 0; r < 9; ++r) s += xv[r] * cc[l * 9 + r];
        mb[e * WSTR + l * 64 + col] = s;
      }
    }
  }
  __syncthreads();
}

// ---------------------------------------------------------------------------
// Pass 1: edge pipeline -> attention logits + segment-max (ordered-uint).
// ---------------------------------------------------------------------------
__global__ void __launch_bounds__(32) k_edge_pass1(
    const float* ea, const float* es, const float* rw1, const float* rb1,
    const float* lnw, const float* lnb, const float* rw2, const float* roff,
    const float* U1, const float* V1, const float* Y1,
    const float* U2, const float* V2, const float* Y2,
    const float* srcF, const float* dstF, const int* esrc, const int* edst,
    const float* Wa, const float* ba, const float* adot,
    float* logit, unsigned* amaxk)
{
  extern __shared__ float smem[];
  float* hb = smem;
  float* wb = hb + 16 * HSTR;
  float* mb = wb + 16 * WSTR;
  float* shb = mb + 16 * WSTR;
  float* ys1b = shb + 192;
  float* cc = ys1b + 192;
  const int lane = threadIdx.x;
  const int eg0 = blockIdx.x * 16;

  edge_common(eg0, lane, ea, es, rw1, rb1, lnw, lnb, rw2, roff,
              U1, V1, Y1, U2, V2, Y2, srcF, dstF, esrc, edst,
              hb, wb, mb, shb, ys1b, cc);

  const int mrow = lane & 15, hi = lane >> 4, koff = hi * 2;
  // alpha = msg'[:,0,:] @ W_alpha + b_alpha (4 WMMA tiles)
  v8f aacc[4] = {};
  for (int k0 = 0; k0 < 64; k0 += 4) {
    v2f a;
    a.x = mb[mrow * WSTR + k0 + koff];
    a.y = mb[mrow * WSTR + k0 + koff + 1];
#pragma unroll
    for (int nt = 0; nt < 4; ++nt) {
      v2f b;
      b.x = Wa[(size_t)(k0 + koff) * 64 + nt * 16 + mrow];
      b.y = Wa[(size_t)(k0 + koff + 1) * 64 + nt * 16 + mrow];
      aacc[nt] = wmma4(a, b, aacc[nt]);
    }
  }
  __syncthreads();
#pragma unroll
  for (int nt = 0; nt < 4; ++nt)
#pragma unroll
    for (int g = 0; g < 8; ++g) {
      int row = g + hi * 8, col = nt * 16 + mrow;
      float x = aacc[nt][g] + ba[col];
      float sg = 1.f / (1.f + expf(-x));
      float y = 0.6f * x + 0.4f * x * (2.f * sg - 1.f);   // smooth_lrelu
      hb[row * HSTR + col] = y * adot[col];
    }
  __syncthreads();
  for (int i = lane; i < 128; i += 32) {
    int e = i >> 3, h = i & 7;
    float s = 0.f;
#pragma unroll
    for (int d = 0; d < 8; ++d) s += hb[e * HSTR + h * 8 + d];
    int de = edst[eg0 + e];
    logit[(size_t)(eg0 + e) * HH + h] = s;
    unsigned u = __float_as_uint(s);
    unsigned key = (u & 0x80000000u) ? ~u : (u | 0x80000000u); // order-preserving
    atomicMax(&amaxk[(size_t)de * HH + h], key);
  }
}

// ---------------------------------------------------------------------------
// Softmax denominator pass.
// ---------------------------------------------------------------------------
__global__ void __launch_bounds__(256) k_soft(
    const float* __restrict__ logit, const unsigned* __restrict__ amaxk,
    const int* __restrict__ edst, float* __restrict__ aexp,
    float* __restrict__ asum)
{
  int i = blockIdx.x * blockDim.x + threadIdx.x;
  if (i >= EE * HH) return;
  int e = i >> 3, h = i & 7;
  int de = edst[e];
  unsigned key = amaxk[(size_t)de * HH + h];
  unsigned u = (key & 0x80000000u) ? (key ^ 0x80000000u) : ~key;
  float amax = __uint_as_float(u);
  float v = expf(logit[i] - amax);
  aexp[i] = v;
  atomicAdd(&asum[(size_t)de * HH + h], v);
}

// ---------------------------------------------------------------------------
// Pass 2: recompute edge pipeline, value GEMM + gating, CP-TP2, attention
// scale, scatter-add into node accumulator.
// ---------------------------------------------------------------------------
__global__ void __launch_bounds__(32) k_edge_pass2(
    const float* ea, const float* es, const float* rw1, const float* rb1,
    const float* lnw, const float* lnb, const float* rw2, const float* roff,
    const float* U1, const float* V1, const float* Y1,
    const float* U2, const float* V2, const float* Y2,
    const float* srcF, const float* dstF, const int* esrc, const int* edst,
    const float* Wv, const float* bv,
    const float* aexp, const float* asum, float* outacc)
{
  extern __shared__ float smem[];
  float* hb = smem;
  float* wb = hb + 16 * HSTR;
  float* mb = wb + 16 * WSTR;
  float* shb = mb + 16 * WSTR;
  float* ys1b = shb + 192;
  float* cc = ys1b + 192;
  float* ys2b = cc + 486;
  float* gb = ys2b + 192;
  float* xb = gb + 16 * HSTR;
  const int lane = threadIdx.x;
  const int eg0 = blockIdx.x * 16;

  edge_common(eg0, lane, ea, es, rw1, rb1, lnw, lnb, rw2, roff,
              U1, V1, Y1, U2, V2, Y2, srcF, dstF, esrc, edst,
              hb, wb, mb, shb, ys1b, cc);

  const int mrow = lane & 15, hi = lane >> 4, koff = hi * 2;
  for (int i = lane; i < 16 * WSTR; i += 32) xb[i] = 0.f;   // xv2 accumulator
  for (int i = lane; i < 144; i += 32) {                    // ys2
    int e = i / 9, r = i % 9;
    float s = 0.f;
#pragma unroll
    for (int l = 0; l < 9; ++l) s += shb[e * 12 + l] * cc[405 + l * 9 + r];
    ys2b[e * 12 + r] = s;
  }
  __syncthreads();

  // value GEMM per l: l==0 -> 128 cols (val+gate), l>=1 -> 64 cols (val only)
  for (int l = 0; l < LL; ++l) {
    int ntiles = (l == 0) ? 8 : 4;
    for (int nt = 0; nt < ntiles; ++nt) {
      v8f acc = {};
      for (int k0 = 0; k0 < 64; k0 += 4) {
        v2f a, b;
        a.x = mb[mrow * WSTR + l * 64 + k0 + koff];
        a.y = mb[mrow * WSTR + l * 64 + k0 + koff + 1];
        b.x = Wv[(size_t)(k0 + koff) * 128 + nt * 16 + mrow];
        b.y = Wv[(size_t)(k0 + koff + 1) * 128 + nt * 16 + mrow];
        acc = wmma4(a, b, acc);
      }
#pragma unroll
      for (int g = 0; g < 8; ++g) {
        int e = g + hi * 8, col = nt * 16 + mrow;
        float x = acc[g] + ((l == 0) ? bv[col] : 0.f);   // bias only on l==0
        if (l == 0 && nt >= 4) {
          gb[e * HSTR + (col - 64)] = 1.f / (1.f + expf(-x));   // gate
        } else {
          float v = (l == 0) ? x / (1.f + expf(-x))             // silu row 0
                             : x * gb[e * HSTR + col];          // gated rows
#pragma unroll
          for (int r = 0; r < 9; ++r)
            xb[e * WSTR + r * 64 + col] += v * cc[324 + l * 9 + r];  // xv2 += v*V2
        }
      }
    }
    if (l == 0) __syncthreads();
  }
  __syncthreads();

  // CP-TP2 + attention scale + scatter-add
  {
    int e = mrow, cb = hi * 32;
    int de = edst[eg0 + e];
    float an[8];
#pragma unroll
    for (int h = 0; h < 8; ++h)
      an[h] = aexp[(size_t)(eg0 + e) * HH + h] /
              (asum[(size_t)de * HH + h] + 1e-16f);
    for (int c = 0; c < 32; ++c) {
      int col = cb + c;
      float t[9];
#pragma unroll
      for (int r = 0; r < 9; ++r)
        t[r] = xb[e * WSTR + r * 64 + col] * ys2b[e * 12 + r] *
               wb[e * WSTR + r * 64 + col];
      float sc = an[col >> 3];
#pragma unroll
      for (int l = 0; l < 9; ++l) {
        float s = 0.f;
#pragma unroll
        for (int r = 0; r < 9; ++r) s += t[r] * cc[243 + l * 9 + r];
        atomicAdd(&outacc[(size_t)de * WN + l * 64 + col], s * sc);
      }
    }
  }
}

// ---------------------------------------------------------------------------
// Kernel 5: out = out_accum @ W_proj (+b_proj on l==0 rows)
// ---------------------------------------------------------------------------
__global__ void __launch_bounds__(128) k_proj(
    const float* __restrict__ X, const float* __restrict__ W,
    const float* __restrict__ b, float* __restrict__ out)
{
  __shared__ float At[16 * HSTR];
  const int t = threadIdx.x;
  const int wave = t >> 5, lane = t & 31;
  const int m0 = blockIdx.x * 16;
  for (int i = t; i < 256; i += 128) {
    float4 v = ((const float4*)(X + (size_t)m0 * 64))[i];
    int row = (i * 4) >> 6, col = (i * 4) & 63;
    float* p = &At[row * HSTR + col];
    p[0] = v.x; p[1] = v.y; p[2] = v.z; p[3] = v.w;
  }
  __syncthreads();
  const int n0 = wave * 16;
  const int mrow = lane & 15, hi = lane >> 4, koff = hi * 2;
  v8f acc = {};
  for (int k0 = 0; k0 < 64; k0 += 4) {
    v2f a, bb;
    a.x = At[mrow * HSTR + k0 + koff];
    a.y = At[mrow * HSTR + k0 + koff + 1];
    bb.x = W[(size_t)(k0 + koff) * 64 + n0 + mrow];
    bb.y = W[(size_t)(k0 + koff + 1) * 64 + n0 + mrow];
    acc = wmma4(a, bb, acc);
  }
#pragma unroll
  for (int g = 0; g < 8; ++g) {
    int row = m0 + g + hi * 8;
    int col = n0 + mrow;
    float v = acc[g];
    if ((row % LL) == 0) v += b[col];
    out[(size_t)row * 64 + col] = v;
  }
}

// ---------------------------------------------------------------------------
extern "C" void kernel_launch(void* const* d_in, const int* in_sizes, int n_in,
                              void* d_out, int out_size, void* d_ws, size_t ws_size,
                              hipStream_t stream) {
  const float* node_input = (const float*)d_in[0];
  const float* edge_attr  = (const float*)d_in[1];
  const float* edge_scal  = (const float*)d_in[2];
  const float* W_src      = (const float*)d_in[3];
  const float* b_src      = (const float*)d_in[4];
  const float* W_dst      = (const float*)d_in[5];
  const float* rad_w1     = (const float*)d_in[6];
  const float* rad_b1     = (const float*)d_in[7];
  const float* rad_ln_w   = (const float*)d_in[8];
  const float* rad_ln_b   = (const float*)d_in[9];
  const float* rad_w2     = (const float*)d_in[10];
  const float* rad_off    = (const float*)d_in[11];
  const float* U1 = (const float*)d_in[12];
  const float* V1 = (const float*)d_in[13];
  const float* Y1 = (const float*)d_in[14];
  const float* U2 = (const float*)d_in[15];
  const float* V2 = (const float*)d_in[16];
  const float* Y2 = (const float*)d_in[17];
  const float* W_alpha = (const float*)d_in[18];
  const float* b_alpha = (const float*)d_in[19];
  const float* adot    = (const float*)d_in[20];
  const float* W_value = (const float*)d_in[21];
  const float* b_value = (const float*)d_in[22];
  const float* W_proj  = (const float*)d_in[23];
  const float* b_proj  = (const float*)d_in[24];
  const int* edge_src  = (const int*)d_in[25];
  const int* edge_dst  = (const int*)d_in[26];
  float* out = (float*)d_out;

  // workspace carve (~77 MB)
  float* ws = (float*)d_ws;
  float* srcF  = ws;
  float* dstF  = srcF + (size_t)NL * 64;
  float* logit = dstF + (size_t)NL * 64;
  float* aexp  = logit + (size_t)EE * HH;
  unsigned* amaxk = (unsigned*)(aexp + (size_t)EE * HH);
  float* asum  = (float*)(amaxk + (size_t)NN * HH);
  float* outacc = asum + (size_t)NN * HH;

  hipMemsetAsync(amaxk, 0, (size_t)NN * HH * 4, stream);
  hipMemsetAsync(asum, 0, (size_t)NN * HH * 4, stream);
  hipMemsetAsync(outacc, 0, (size_t)NL * 64 * 4, stream);

  k_node_lin<<<NL / 16, 256, 0, stream>>>(node_input, W_src, b_src, W_dst,
                                          srcF, dstF);

  const size_t SH1 = (16 * HSTR + 2 * 16 * WSTR + 192 + 192 + 486) * sizeof(float);
  const size_t SH2 = SH1 + (192 + 16 * HSTR + 16 * WSTR) * sizeof(float);

  k_edge_pass1<<<EE / 16, 32, SH1, stream>>>(
      edge_attr, edge_scal, rad_w1, rad_b1, rad_ln_w, rad_ln_b, rad_w2,
      rad_off, U1, V1, Y1, U2, V2, Y2, srcF, dstF, edge_src, edge_dst,
      W_alpha, b_alpha, adot, logit, amaxk);

  k_soft<<<(EE * HH + 255) / 256, 256, 0, stream>>>(logit, amaxk, edge_dst,
                                                    aexp, asum);

  k_edge_pass2<<<EE / 16, 32, SH2, stream>>>(
      edge_attr, edge_scal, rad_w1, rad_b1, rad_ln_w, rad_ln_b, rad_w2,
      rad_off, U1, V1, Y1, U2, V2, Y2, srcF, dstF, edge_src, edge_dst,
      W_value, b_value, aexp, asum, outacc);

  k_proj<<<NL / 16, 128, 0, stream>>>(outacc, W_proj, b_proj, out);
}